// UnrolledSolver_71339406786806
// MI455X (gfx1250) — compile-verified
//
#include <hip/hip_runtime.h>
#include <math.h>

typedef float v2f __attribute__((ext_vector_type(2)));
typedef float v8f __attribute__((ext_vector_type(8)));

__device__ inline v8f v8f_zero() {
  v8f z = {0.f,0.f,0.f,0.f,0.f,0.f,0.f,0.f};
  return z;
}

__device__ inline v8f wmma_f32(v2f a, v2f b, v8f c) {
  // D = A(16x4 f32) x B(4x16 f32) + C(16x16 f32)
  return __builtin_amdgcn_wmma_f32_16x16x4_f32(false, a, false, b, (short)0, c, false, false);
}

// ---------------------------------------------------------------- init x = [va; vm]
__global__ void k_init_x(const float* __restrict__ va, const float* __restrict__ vm,
                         float* __restrict__ x, int n) {
  int i = blockIdx.x * blockDim.x + threadIdx.x;
  if (i < n) { x[i] = va[i]; x[n + i] = vm[i]; }
}

// ---------------------------------------------------------------- edge scatter: P/Q + Jacobian
__global__ void k_edge(const float* __restrict__ x, const int* __restrict__ ei,
                       const int* __restrict__ bt,
                       const float* __restrict__ br_r, const float* __restrict__ br_x,
                       const float* __restrict__ b_fr, const float* __restrict__ b_to,
                       const float* __restrict__ tap, const float* __restrict__ shift,
                       float* __restrict__ P, float* __restrict__ Q,
                       float* __restrict__ J, int n, int n2, int E) {
  int e = blockIdx.x * blockDim.x + threadIdx.x;
  if (e >= E) return;
  int s = ei[e], d = ei[E + e];
  float r = br_r[e], xr = br_x[e];
  float den = r * r + xr * xr;
  float Yr = r / den, Yx = -xr / den;
  float va_s = x[s], va_d = x[d];
  float v_a = x[n + s], v_b = x[n + d];
  float tp = tap[e], sh = shift[e];
  float dtf = va_s - va_d;
  float cf = cosf(dtf - sh),  sf_ = sinf(dtf - sh);
  float cr = cosf(-dtf - sh), sr  = sinf(-dtf - sh);
  float bl = b_fr[e] + b_to[e];
  float vab = v_a * v_b / tp;
  float via = v_a / tp;
  float Pfs = vab * (-Yr*cf - Yx*sf_) + Yr * via * via;
  float Pfd = vab * (-Yr*cr - Yx*sr)  + Yr * v_b * v_b;
  float Qfs = vab * (-Yr*sf_ + Yx*cf) - (Yx + bl*0.5f) * via * via;
  float Qfd = vab * (-Yr*sr + Yx*cr)  - (Yx + bl*0.5f) * v_b * v_b;
  atomicAdd(&P[s], Pfs); atomicAdd(&P[d], Pfd);
  atomicAdd(&Q[s], Qfs); atomicAdd(&Q[d], Qfd);

  float dPfs_tha = vab * (Yr*sf_ - Yx*cf);
  float dPfs_thb = -dPfs_tha;
  float dPfs_va  = v_b/tp * (-Yr*cf - Yx*sf_) + 2.f*Yr*v_a/(tp*tp);
  float dPfs_vb  = v_a/tp * (-Yr*cf - Yx*sf_);
  float dPfd_thb = vab * (Yr*sr - Yx*cr);
  float dPfd_tha = -dPfd_thb;
  float dPfd_vb  = v_a/tp * (-Yr*cr - Yx*sr) + 2.f*Yr*v_b;
  float dPfd_va  = v_b/tp * (-Yr*cr - Yx*sr);
  float dQfs_tha = vab * (-Yr*cf - Yx*sf_);
  float dQfs_thb = -dQfs_tha;
  float dQfs_va  = v_b/tp * (-Yr*sf_ + Yx*cf) - 2.f*(Yx + bl*0.5f)*v_a/(tp*tp);
  float dQfs_vb  = v_a/tp * (-Yr*sf_ + Yx*cf);
  float dQfd_thb = vab * (-Yr*cr - Yx*sr);
  float dQfd_tha = -dQfd_thb;
  float dQfd_vb  = v_a/tp * (-Yr*sr + Yx*cr) - 2.f*(Yx + bl*0.5f)*v_b;
  float dQfd_va  = v_b/tp * (-Yr*sr + Yx*cr);

  int bts = bt[s], btd = bt[d];
  float kPs = (bts == 3) ? 0.f : 1.f;
  float kPd = (btd == 3) ? 0.f : 1.f;
  float kQs = (bts == 2 || bts == 3) ? 0.f : 1.f;
  float kQd = (btd == 2 || btd == 3) ? 0.f : 1.f;
  size_t N2 = (size_t)n2;
  // dFP/dva
  atomicAdd(&J[(size_t)s*N2 + s], -dPfs_tha * kPs);
  atomicAdd(&J[(size_t)s*N2 + d], -dPfs_thb * kPs);
  atomicAdd(&J[(size_t)d*N2 + d], -dPfd_thb * kPd);
  atomicAdd(&J[(size_t)d*N2 + s], -dPfd_tha * kPd);
  // dFP/dvm
  atomicAdd(&J[(size_t)s*N2 + n + s], -dPfs_va * kPs);
  atomicAdd(&J[(size_t)s*N2 + n + d], -dPfs_vb * kPs);
  atomicAdd(&J[(size_t)d*N2 + n + d], -dPfd_vb * kPd);
  atomicAdd(&J[(size_t)d*N2 + n + s], -dPfd_va * kPd);
  // dFQ/dva
  atomicAdd(&J[(size_t)(n+s)*N2 + s], -dQfs_tha * kQs);
  atomicAdd(&J[(size_t)(n+s)*N2 + d], -dQfs_thb * kQs);
  atomicAdd(&J[(size_t)(n+d)*N2 + d], -dQfd_thb * kQd);
  atomicAdd(&J[(size_t)(n+d)*N2 + s], -dQfd_tha * kQd);
  // dFQ/dvm
  atomicAdd(&J[(size_t)(n+s)*N2 + n + s], -dQfs_va * kQs);
  atomicAdd(&J[(size_t)(n+s)*N2 + n + d], -dQfs_vb * kQs);
  atomicAdd(&J[(size_t)(n+d)*N2 + n + d], -dQfd_vb * kQd);
  atomicAdd(&J[(size_t)(n+d)*N2 + n + s], -dQfd_va * kQd);
}

// ---------------------------------------------------------------- per-bus: F assembly + J diag terms
__global__ void k_bus(const float* __restrict__ x,
                      const float* __restrict__ gs, const float* __restrict__ bs,
                      const float* __restrict__ p_spec, const float* __restrict__ q_spec,
                      const float* __restrict__ vm_sp, const int* __restrict__ bt,
                      const float* __restrict__ P, const float* __restrict__ Q,
                      float* __restrict__ F, float* __restrict__ J, int n, int n2) {
  int i = blockIdx.x * blockDim.x + threadIdx.x;
  if (i >= n) return;
  float va = x[i], vm = x[n + i];
  int b = bt[i];
  bool sl = (b == 3), pv = (b == 2);
  float Pt = P[i] + vm * vm * gs[i];
  float Qt = Q[i] - vm * vm * bs[i];
  F[i]     = sl          ? va              : (p_spec[i] - Pt);
  F[n + i] = (pv || sl)  ? (vm - vm_sp[i]) : (q_spec[i] - Qt);
  float kP = sl ? 0.f : 1.f;
  float kQ = (pv || sl) ? 0.f : 1.f;
  size_t N2 = (size_t)n2;
  atomicAdd(&J[(size_t)i*N2 + n + i],      -2.f * vm * gs[i] * kP);
  atomicAdd(&J[(size_t)(n+i)*N2 + n + i],   2.f * vm * bs[i] * kQ);
  if (sl)       atomicAdd(&J[(size_t)i*N2 + i], 1.f);
  if (pv || sl) atomicAdd(&J[(size_t)(n+i)*N2 + n + i], 1.f);
}

// ---------------------------------------------------------------- H = J^T J + diag(d) + (U*s)U^T via f32 WMMA
// block = 256 threads (8 waves); block tile 64(rows i) x 128(cols j); wave tile 32x32 = 2x2 WMMA accs
// LDS tiles are column-major [col][k] with even pad so each lane's (K,K+1) operand
// pair is one aligned ds_load_b64 (no register shuffles), conflict-free banking.
#define SYRK_BK 32
__global__ __launch_bounds__(256) void k_syrk_wmma(const float* __restrict__ J,
                                                   const float* __restrict__ dvec,
                                                   const float* __restrict__ U,
                                                   const float* __restrict__ svec,
                                                   float* __restrict__ H, int n2, int Kr) {
  __shared__ alignas(16) float As[64][SYRK_BK + 2];    // [i-col][k], stride 34 (even)
  __shared__ alignas(16) float Bs[128][SYRK_BK + 2];   // [j-col][k]
  int tid = threadIdx.x;
  int lane = tid & 31;
  int wave = tid >> 5;
  int wrow = wave & 1;       // 0..1
  int wcol = wave >> 1;      // 0..3
  int i0 = blockIdx.y * 64;
  int j0 = blockIdx.x * 128;
  v8f acc[2][2];
  acc[0][0] = v8f_zero(); acc[0][1] = v8f_zero();
  acc[1][0] = v8f_zero(); acc[1][1] = v8f_zero();
  int m  = lane & 15;
  int kh = (lane >> 4) * 2;

  for (int k0 = 0; k0 < n2; k0 += SYRK_BK) {
    for (int t = tid; t < SYRK_BK * 64; t += 256) {
      int kk = t >> 6, ii = t & 63;
      As[ii][kk] = J[(size_t)(k0 + kk) * n2 + i0 + ii];
      if (k0 + SYRK_BK < n2)
        __builtin_prefetch(&J[(size_t)(k0 + SYRK_BK + kk) * n2 + i0 + ii], 0, 3);
    }
    for (int t = tid; t < SYRK_BK * 128; t += 256) {
      int kk = t >> 7, jj = t & 127;
      Bs[jj][kk] = J[(size_t)(k0 + kk) * n2 + j0 + jj];
      if (k0 + SYRK_BK < n2)
        __builtin_prefetch(&J[(size_t)(k0 + SYRK_BK + kk) * n2 + j0 + jj], 0, 3);
    }
    __syncthreads();
#pragma unroll
    for (int kk = 0; kk < SYRK_BK; kk += 4) {
      v2f a[2], b[2];
#pragma unroll
      for (int tr = 0; tr < 2; tr++)
        a[tr] = *(const v2f*)&As[wrow * 32 + tr * 16 + m][kk + kh];
#pragma unroll
      for (int tc = 0; tc < 2; tc++)
        b[tc] = *(const v2f*)&Bs[wcol * 32 + tc * 16 + m][kk + kh];
#pragma unroll
      for (int tr = 0; tr < 2; tr++)
#pragma unroll
        for (int tc = 0; tc < 2; tc++)
          acc[tr][tc] = wmma_f32(a[tr], b[tc], acc[tr][tc]);
    }
    __syncthreads();
  }

  // epilogue: + diag(d) + rank-K (U*s)U^T, store H
#pragma unroll
  for (int tr = 0; tr < 2; tr++) {
#pragma unroll
    for (int tc = 0; tc < 2; tc++) {
#pragma unroll
      for (int c = 0; c < 8; c++) {
        int mi = c + 8 * (lane >> 4);
        int ni = lane & 15;
        int gi = i0 + wrow * 32 + tr * 16 + mi;
        int gj = j0 + wcol * 32 + tc * 16 + ni;
        float v = acc[tr][tc][c];
        if (gi == gj) v += dvec[gi];
        float dot = 0.f;
        for (int t = 0; t < Kr; t++)
          dot += U[(size_t)gi * Kr + t] * svec[t] * U[(size_t)gj * Kr + t];
        H[(size_t)gi * n2 + gj] = v + dot;
      }
    }
  }
}

// ---------------------------------------------------------------- g = -(J^T F)   (coalesced across columns)
__global__ void k_gemv_jt(const float* __restrict__ J, const float* __restrict__ F,
                          float* __restrict__ g, int n2) {
  int c = blockIdx.x * blockDim.x + threadIdx.x;
  if (c >= n2) return;
  float s = 0.f;
  for (int r = 0; r < n2; r++) s += J[(size_t)r * n2 + c] * F[r];
  g[c] = -s;
}

// ---------------------------------------------------------------- blocked Cholesky (NB=32), L mirrored into upper
__global__ __launch_bounds__(32) void k_chol_diag(float* __restrict__ H, int n2, int pc) {
  __shared__ float A[32][33];
  int l = threadIdx.x;
  for (int r = 0; r < 32; r++) A[r][l] = H[(size_t)(pc + r) * n2 + pc + l];
  __syncthreads();
  for (int c = 0; c < 32; c++) {
    if (l == 0) A[c][c] = sqrtf(A[c][c]);
    __syncthreads();
    if (l > c) A[l][c] /= A[c][c];
    __syncthreads();
    if (l > c) {
      float lc = A[l][c];
      for (int t = c + 1; t <= l; t++) A[l][t] -= lc * A[t][c];
    }
    __syncthreads();
  }
  for (int r = 0; r < 32; r++)
    H[(size_t)(pc + r) * n2 + pc + l] = (l <= r) ? A[r][l] : A[l][r];  // L and L^T
}

__global__ __launch_bounds__(256) void k_chol_panel(float* __restrict__ H, int n2, int pc) {
  __shared__ float L11[32][33];
  int tid = threadIdx.x;
  for (int t = tid; t < 32 * 32; t += 256) {
    int r = t >> 5, c = t & 31;
    L11[r][c] = H[(size_t)(pc + r) * n2 + pc + c];
  }
  __syncthreads();
  int to = pc + 32;
  int r = to + blockIdx.x * 256 + tid;
  if (r >= n2) return;
  float Lr[32];
  for (int c = 0; c < 32; c++) {
    float s = H[(size_t)r * n2 + pc + c];
    for (int t = 0; t < c; t++) s -= Lr[t] * L11[c][t];
    Lr[c] = s / L11[c][c];
  }
  for (int c = 0; c < 32; c++) {
    H[(size_t)r * n2 + pc + c] = Lr[c];
    H[(size_t)(pc + c) * n2 + r] = Lr[c];  // mirror for coalesced trsv
  }
}

// trailing update T -= L21 * L21^T  (K=32, f32 WMMA; lower block-triangle only)
__global__ __launch_bounds__(32) void k_chol_update(float* __restrict__ H, int n2, int pc) {
  if (blockIdx.y < blockIdx.x) return;  // only lower blocks (incl. diagonal)
  int to = pc + 32;
  int lane = threadIdx.x;
  int i0 = to + blockIdx.y * 32;
  int j0 = to + blockIdx.x * 32;
  __shared__ alignas(16) float Ls[2][32][34];  // even pad -> aligned b64 operand loads
  for (int t = lane; t < 32 * 32; t += 32) {
    int r = t >> 5, c = t & 31;
    Ls[0][r][c] = H[(size_t)(i0 + r) * n2 + pc + c];
    Ls[1][r][c] = H[(size_t)(j0 + r) * n2 + pc + c];
  }
  __syncthreads();
  v8f acc[2][2];
  acc[0][0] = v8f_zero(); acc[0][1] = v8f_zero();
  acc[1][0] = v8f_zero(); acc[1][1] = v8f_zero();
  int m  = lane & 15;
  int kh = (lane >> 4) * 2;
#pragma unroll
  for (int kk = 0; kk < 32; kk += 4) {
    v2f a[2], b[2];
#pragma unroll
    for (int tr = 0; tr < 2; tr++)
      a[tr] = *(const v2f*)&Ls[0][tr * 16 + m][kk + kh];
#pragma unroll
    for (int tc = 0; tc < 2; tc++)
      b[tc] = *(const v2f*)&Ls[1][tc * 16 + m][kk + kh];
#pragma unroll
    for (int tr = 0; tr < 2; tr++)
#pragma unroll
      for (int tc = 0; tc < 2; tc++)
        acc[tr][tc] = wmma_f32(a[tr], b[tc], acc[tr][tc]);
  }
#pragma unroll
  for (int tr = 0; tr < 2; tr++)
#pragma unroll
    for (int tc = 0; tc < 2; tc++)
#pragma unroll
      for (int c = 0; c < 8; c++) {
        int gi = i0 + tr * 16 + c + 8 * (lane >> 4);
        int gj = j0 + tc * 16 + (lane & 15);
        H[(size_t)gi * n2 + gj] -= acc[tr][tc][c];
      }
}

// ---------------------------------------------------------------- triangular solves (in place on g)
__global__ __launch_bounds__(1024) void k_trsv_fwd(const float* __restrict__ H,
                                                   float* __restrict__ g, int n2) {
  int tid = threadIdx.x;
  for (int j = 0; j < n2; j++) {
    __syncthreads();
    if (tid == 0) g[j] = g[j] / H[(size_t)j * n2 + j];
    __syncthreads();
    float yj = g[j];
    for (int r = j + 1 + tid; r < n2; r += 1024)
      g[r] -= H[(size_t)j * n2 + r] * yj;  // L[r,j] via upper mirror -> coalesced
  }
}

__global__ __launch_bounds__(1024) void k_trsv_bwd(const float* __restrict__ H,
                                                   float* __restrict__ g, int n2) {
  int tid = threadIdx.x;
  for (int j = n2 - 1; j >= 0; j--) {
    __syncthreads();
    if (tid == 0) g[j] = g[j] / H[(size_t)j * n2 + j];
    __syncthreads();
    float dj = g[j];
    for (int r = tid; r < j; r += 1024)
      g[r] -= H[(size_t)j * n2 + r] * dj;  // L[j,r] lower row j -> coalesced
  }
}

__global__ void k_axpy(float* __restrict__ x, const float* __restrict__ dx, int n2) {
  int i = blockIdx.x * blockDim.x + threadIdx.x;
  if (i < n2) x[i] += dx[i];
}

__global__ void k_copy_out(float* __restrict__ out, const float* __restrict__ x, int n2) {
  int i = blockIdx.x * blockDim.x + threadIdx.x;
  if (i < n2) out[i] = x[i];
}

// ================================================================ host
extern "C" void kernel_launch(void* const* d_in, const int* in_sizes, int n_in,
                              void* d_out, int out_size, void* d_ws, size_t ws_size,
                              hipStream_t stream) {
  const float* va    = (const float*)d_in[0];
  const float* vm    = (const float*)d_in[1];
  const float* br_r  = (const float*)d_in[2];
  const float* br_x  = (const float*)d_in[3];
  const float* b_fr  = (const float*)d_in[4];
  const float* b_to  = (const float*)d_in[5];
  const float* tap   = (const float*)d_in[6];
  const float* shift = (const float*)d_in[7];
  const float* p_spec= (const float*)d_in[8];
  const float* q_spec= (const float*)d_in[9];
  const float* gs    = (const float*)d_in[10];
  const float* bs    = (const float*)d_in[11];
  const float* vm_sp = (const float*)d_in[12];
  const float* dvec  = (const float*)d_in[13];
  const float* U     = (const float*)d_in[14];
  const float* svec  = (const float*)d_in[15];
  const int*   ei    = (const int*)d_in[16];
  const int*   bt    = (const int*)d_in[17];

  int n  = in_sizes[0];
  int E  = in_sizes[2];
  int Kr = in_sizes[15];
  int n2 = 2 * n;

  float* ws = (float*)d_ws;
  size_t mat = (size_t)n2 * (size_t)n2;
  float* J = ws;
  float* H = ws + mat;
  float* x = ws + 2 * mat;
  float* F = x + n2;
  float* P = F + n2;      // P[n] then Q[n] contiguous
  float* Q = P + n;
  float* g = Q + n;

  k_init_x<<<(n + 255) / 256, 256, 0, stream>>>(va, vm, x, n);

  const int N_STEPS = 2;
  for (int step = 0; step < N_STEPS; step++) {
    hipMemsetAsync(J, 0, mat * sizeof(float), stream);
    hipMemsetAsync(P, 0, (size_t)n2 * sizeof(float), stream);  // P and Q

    k_edge<<<(E + 255) / 256, 256, 0, stream>>>(x, ei, bt, br_r, br_x, b_fr, b_to,
                                                tap, shift, P, Q, J, n, n2, E);
    k_bus<<<(n + 255) / 256, 256, 0, stream>>>(x, gs, bs, p_spec, q_spec, vm_sp,
                                               bt, P, Q, F, J, n, n2);

    dim3 sgrid(n2 / 128, n2 / 64);
    k_syrk_wmma<<<sgrid, 256, 0, stream>>>(J, dvec, U, svec, H, n2, Kr);

    k_gemv_jt<<<(n2 + 255) / 256, 256, 0, stream>>>(J, F, g, n2);

    for (int pc = 0; pc < n2; pc += 32) {
      k_chol_diag<<<1, 32, 0, stream>>>(H, n2, pc);
      int m = n2 - pc - 32;
      if (m > 0) {
        k_chol_panel<<<(m + 255) / 256, 256, 0, stream>>>(H, n2, pc);
        int mb = m / 32;
        dim3 ugrid(mb, mb);
        k_chol_update<<<ugrid, 32, 0, stream>>>(H, n2, pc);
      }
    }

    k_trsv_fwd<<<1, 1024, 0, stream>>>(H, g, n2);
    k_trsv_bwd<<<1, 1024, 0, stream>>>(H, g, n2);
    k_axpy<<<(n2 + 255) / 256, 256, 0, stream>>>(x, g, n2);
  }

  k_copy_out<<<(n2 + 255) / 256, 256, 0, stream>>>((float*)d_out, x, n2);
}